// SplineNet_85590108274955
// MI455X (gfx1250) — compile-verified
//
#include <hip/hip_runtime.h>

typedef __attribute__((ext_vector_type(2))) float v2f;
typedef __attribute__((ext_vector_type(8))) float v8f;

#define WAVES_PER_BLOCK 4

static inline int ceil_div_i(long long a, long long b) { return (int)((a + b - 1) / b); }

// ---------------------------------------------------------------------------
// Dense GEMM via V_WMMA_F32_16X16X4_F32.
// Computes Y[n, which*64 + o] = sum_i X[n,i] * Wsel[i,o]  for which in {0,1,2}
// (blockIdx.y selects W[0], W[1], root). X is [nNodes x 64] row-major,
// W is [64 x OUTD] row-major, Y has row stride 192 floats.
// One wave handles a 16-row tile x OUTD cols. K = 64 always.
// A 16x4 f32 frag: lane L -> A[L%16][2*(L/16)+c] in VGPR c.
// B 4x16 frag (mirror): lane L -> B[2*(L/16)+c][L%16] in VGPR c.
// D 16x16: lane L, VGPR r -> D[r + 8*(L/16)][L%16].
// ---------------------------------------------------------------------------
template <int OUTD>
__global__ __launch_bounds__(128) void spline_gemm(
    const float* __restrict__ X, const float* __restrict__ W0,
    const float* __restrict__ W1, const float* __restrict__ Wr,
    float* __restrict__ Y, int nNodes)
{
    constexpr int NT = (OUTD + 15) / 16;
    const int lane    = threadIdx.x & 31;
    const int wave    = threadIdx.x >> 5;
    const int rowTile = blockIdx.x * WAVES_PER_BLOCK + wave;
    const int rowBase = rowTile * 16;
    if (rowBase >= nNodes) return;           // wave-uniform: EXEC stays all-ones

    const int which = blockIdx.y;
    const float* W  = (which == 0) ? W0 : ((which == 1) ? W1 : Wr);
    const int yoff  = which * 64;

    const int m  = lane & 15;                // row within tile / col within tile
    const int kh = (lane >> 4) << 1;         // K offset 0 or 2
    const int rh = (lane >> 4) << 3;         // D row-half offset 0 or 8

    v8f acc[NT];
#pragma unroll
    for (int j = 0; j < NT; ++j) acc[j] = v8f{};

    int row = rowBase + m;
    if (row >= nNodes) row = nNodes - 1;     // clamp (no divergence)

#pragma unroll 4
    for (int k0 = 0; k0 < 64; k0 += 4) {
        v2f a = *(const v2f*)(X + (size_t)row * 64 + (k0 + kh));
#pragma unroll
        for (int j = 0; j < NT; ++j) {
            const int n = j * 16 + m;
            const bool ok = (OUTD % 16 == 0) || (n < OUTD);
            v2f b;
            b.x = ok ? W[(size_t)(k0 + kh)     * OUTD + n] : 0.0f;
            b.y = ok ? W[(size_t)(k0 + kh + 1) * OUTD + n] : 0.0f;
            acc[j] = __builtin_amdgcn_wmma_f32_16x16x4_f32(
                false, a, false, b, (short)0, acc[j], false, false);
        }
    }

#pragma unroll
    for (int j = 0; j < NT; ++j) {
        const int n = j * 16 + m;
        if ((OUTD % 16 == 0) || (n < OUTD)) {
#pragma unroll
            for (int r = 0; r < 8; ++r) {
                const int orow = rowBase + rh + r;
                if (orow < nNodes)
                    Y[(size_t)orow * 192 + yoff + n] = acc[j][r];
            }
        }
    }
}

// ---------------------------------------------------------------------------
// Edge scatter: msg = (1-t)*Y0[src] + t*Y1[src], atomically accumulated into
// agg[dst]. One thread per (edge, 4-float chunk).
// ---------------------------------------------------------------------------
__global__ void spline_scatter(const float* __restrict__ Y,
                               const float* __restrict__ ea,
                               const int* __restrict__ ei,
                               float* __restrict__ agg,
                               int nEdges, int outDim, int chunks)
{
    long long tid   = (long long)blockIdx.x * blockDim.x + threadIdx.x;
    long long total = (long long)nEdges * chunks;
    if (tid >= total) return;
    const int e = (int)(tid / chunks);
    const int c = (int)(tid % chunks);

    const int src = ei[e];
    const int dst = ei[nEdges + e];
    const float t  = ea[e];
    const float w0 = 1.0f - t;

    const float* y0 = Y + (size_t)src * 192 + c * 4;
    const float4 a = *(const float4*)(y0);
    const float4 b = *(const float4*)(y0 + 64);

    float* d = agg + (size_t)dst * 64 + c * 4;
    const int f = c * 4;
    if (f + 0 < outDim) atomicAdd(d + 0, w0 * a.x + t * b.x);
    if (f + 1 < outDim) atomicAdd(d + 1, w0 * a.y + t * b.y);
    if (f + 2 < outDim) atomicAdd(d + 2, w0 * a.z + t * b.z);
    if (f + 3 < outDim) atomicAdd(d + 3, w0 * a.w + t * b.w);
}

// Degree (edges per destination node) -- layer-invariant, computed once.
__global__ void spline_degree(const int* __restrict__ ei, float* __restrict__ deg,
                              int nEdges)
{
    int e = blockIdx.x * blockDim.x + threadIdx.x;
    if (e >= nEdges) return;
    atomicAdd(&deg[ei[nEdges + e]], 1.0f);
}

// h = agg/max(deg,1) + (X@root) + bias, optional SiLU; write to Xout.
__global__ void spline_finalize(const float* __restrict__ agg,
                                const float* __restrict__ Y,
                                const float* __restrict__ deg,
                                const float* __restrict__ bias,
                                float* __restrict__ Xout,
                                int nNodes, int outDim, int doSilu)
{
    int tid = blockIdx.x * blockDim.x + threadIdx.x;
    if (tid >= nNodes * outDim) return;
    const int n = tid / outDim;
    const int f = tid % outDim;
    const float dinv = 1.0f / fmaxf(deg[n], 1.0f);
    float h = agg[(size_t)n * 64 + f] * dinv + Y[(size_t)n * 192 + 128 + f] + bias[f];
    if (doSilu) h = h / (1.0f + __expf(-h));
    Xout[(size_t)n * 64 + f] = h;
}

// Order-preserving float<->uint encoding for atomicMax on floats.
__device__ __forceinline__ unsigned f2ord(float f) {
    unsigned u = __float_as_uint(f);
    return (u & 0x80000000u) ? ~u : (u | 0x80000000u);
}
__device__ __forceinline__ float ord2f(unsigned u) {
    return (u & 0x80000000u) ? __uint_as_float(u & 0x7FFFFFFFu)
                             : __uint_as_float(~u);
}

// Last layer: compute h (no SiLU) and feed pooling atomics directly.
__global__ void spline_finalize_pool(const float* __restrict__ agg,
                                     const float* __restrict__ Y,
                                     const float* __restrict__ deg,
                                     const float* __restrict__ bias,
                                     const int* __restrict__ batch,
                                     unsigned* __restrict__ gmaxE,
                                     float* __restrict__ gsum,
                                     float* __restrict__ gcnt,
                                     int nNodes, int outDim)
{
    int tid = blockIdx.x * blockDim.x + threadIdx.x;
    if (tid >= nNodes * outDim) return;
    const int n = tid / outDim;
    const int f = tid % outDim;
    const float dinv = 1.0f / fmaxf(deg[n], 1.0f);
    const float h = agg[(size_t)n * 64 + f] * dinv + Y[(size_t)n * 192 + 128 + f] + bias[f];
    const int g = batch[n];
    atomicMax(&gmaxE[g * outDim + f], f2ord(h));
    atomicAdd(&gsum[g * outDim + f], h);
    if (f == 0) atomicAdd(&gcnt[g], 1.0f);
}

// One thread per graph: concat [max | mean], log-softmax over 20 logits.
__global__ void pool_out(const unsigned* __restrict__ gmaxE,
                         const float* __restrict__ gsum,
                         const float* __restrict__ gcnt,
                         float* __restrict__ out, int nGraphs)
{
    int g = blockIdx.x * blockDim.x + threadIdx.x;
    if (g >= nGraphs) return;
    float v[20];
    const float cnt = fmaxf(gcnt[g], 1.0f);
#pragma unroll
    for (int f = 0; f < 10; ++f) {
        unsigned u = gmaxE[g * 10 + f];
        float mx = (u == 0u) ? 0.0f : ord2f(u);
        if ((__float_as_uint(mx) & 0x7F800000u) == 0x7F800000u) mx = 0.0f; // non-finite -> 0
        v[f]      = mx;
        v[10 + f] = gsum[g * 10 + f] / cnt;
    }
    float m = v[0];
#pragma unroll
    for (int i = 1; i < 20; ++i) m = fmaxf(m, v[i]);
    float s = 0.0f;
#pragma unroll
    for (int i = 0; i < 20; ++i) s += __expf(v[i] - m);
    const float lse = m + __logf(s);
#pragma unroll
    for (int i = 0; i < 20; ++i) out[g * 20 + i] = v[i] - lse;
}

__global__ void fill_f32(float* __restrict__ p, float v, long long n)
{
    long long i = (long long)blockIdx.x * blockDim.x + threadIdx.x;
    if (i < n) p[i] = v;
}

// ---------------------------------------------------------------------------
extern "C" void kernel_launch(void* const* d_in, const int* in_sizes, int n_in,
                              void* d_out, int out_size, void* d_ws, size_t ws_size,
                              hipStream_t stream)
{
    const float* x     = (const float*)d_in[0];
    const int*   ei    = (const int*)d_in[1];
    const float* ea    = (const float*)d_in[2];
    const int*   batch = (const int*)d_in[3];
    const float* W0 = (const float*)d_in[4];
    const float* r0 = (const float*)d_in[5];
    const float* b0 = (const float*)d_in[6];
    const float* W1 = (const float*)d_in[7];
    const float* r1 = (const float*)d_in[8];
    const float* b1 = (const float*)d_in[9];
    const float* W2 = (const float*)d_in[10];
    const float* r2 = (const float*)d_in[11];
    const float* b2 = (const float*)d_in[12];
    float* out = (float*)d_out;

    const int nNodes  = in_sizes[0] / 64;   // 50000
    const int nEdges  = in_sizes[1] / 2;    // 800000
    const int nGraphs = 64;
    (void)n_in; (void)out_size; (void)ws_size;

    // Workspace carve-out (256B aligned)
    char* ws = (char*)d_ws;
    size_t off = 0;
    auto carve = [&](size_t bytes) -> void* {
        void* p = ws + off;
        off += (bytes + 255) & ~(size_t)255;
        return p;
    };
    float*    Y     = (float*)carve((size_t)nNodes * 192 * sizeof(float)); // [Y0|Y1|R]
    float*    agg   = (float*)carve((size_t)nNodes * 64 * sizeof(float));
    float*    X0    = (float*)carve((size_t)nNodes * 64 * sizeof(float));
    float*    X1    = (float*)carve((size_t)nNodes * 64 * sizeof(float));
    float*    deg   = (float*)carve((size_t)nNodes * sizeof(float));
    unsigned* gmaxE = (unsigned*)carve((size_t)nGraphs * 10 * sizeof(unsigned));
    float*    gsum  = (float*)carve((size_t)nGraphs * 10 * sizeof(float));
    float*    gcnt  = (float*)carve((size_t)nGraphs * sizeof(float));

    const int BLK = 256;
    const long long aggN = (long long)nNodes * 64;

    // Init degree + pooling accumulators
    fill_f32<<<ceil_div_i(nNodes, BLK), BLK, 0, stream>>>(deg, 0.0f, nNodes);
    fill_f32<<<1, BLK, 0, stream>>>((float*)gmaxE, 0.0f, nGraphs * 10); // 0u == -inf sentinel
    fill_f32<<<1, BLK, 0, stream>>>(gsum, 0.0f, nGraphs * 10);
    fill_f32<<<1, BLK, 0, stream>>>(gcnt, 0.0f, nGraphs);
    spline_degree<<<ceil_div_i(nEdges, BLK), BLK, 0, stream>>>(ei, deg, nEdges);

    const int rowTiles = (nNodes + 15) / 16;
    dim3 ggrid(ceil_div_i(rowTiles, WAVES_PER_BLOCK), 3);

    // ---- Layer 0: x -> X0 (out 64, SiLU) ----
    fill_f32<<<ceil_div_i(aggN, BLK), BLK, 0, stream>>>(agg, 0.0f, aggN);
    spline_gemm<64><<<ggrid, 128, 0, stream>>>(x, W0, W0 + 64 * 64, r0, Y, nNodes);
    spline_scatter<<<ceil_div_i((long long)nEdges * 16, BLK), BLK, 0, stream>>>(
        Y, ea, ei, agg, nEdges, 64, 16);
    spline_finalize<<<ceil_div_i(aggN, BLK), BLK, 0, stream>>>(
        agg, Y, deg, b0, X0, nNodes, 64, 1);

    // ---- Layer 1: X0 -> X1 (out 64, SiLU) ----
    fill_f32<<<ceil_div_i(aggN, BLK), BLK, 0, stream>>>(agg, 0.0f, aggN);
    spline_gemm<64><<<ggrid, 128, 0, stream>>>(X0, W1, W1 + 64 * 64, r1, Y, nNodes);
    spline_scatter<<<ceil_div_i((long long)nEdges * 16, BLK), BLK, 0, stream>>>(
        Y, ea, ei, agg, nEdges, 64, 16);
    spline_finalize<<<ceil_div_i(aggN, BLK), BLK, 0, stream>>>(
        agg, Y, deg, b1, X1, nNodes, 64, 1);

    // ---- Layer 2: X1 -> pooling (out 10, no SiLU) ----
    fill_f32<<<ceil_div_i(aggN, BLK), BLK, 0, stream>>>(agg, 0.0f, aggN);
    spline_gemm<10><<<ggrid, 128, 0, stream>>>(X1, W2, W2 + 64 * 10, r2, Y, nNodes);
    spline_scatter<<<ceil_div_i((long long)nEdges * 3, BLK), BLK, 0, stream>>>(
        Y, ea, ei, agg, nEdges, 10, 3);
    spline_finalize_pool<<<ceil_div_i((long long)nNodes * 10, BLK), BLK, 0, stream>>>(
        agg, Y, deg, b2, batch, gmaxE, gsum, gcnt, nNodes, 10);

    // ---- Readout: [max | mean] -> log_softmax ----
    pool_out<<<1, 64, 0, stream>>>(gmaxE, gsum, gcnt, out, nGraphs);
}